// CoTrackerDensePredictor_67216238182574
// MI455X (gfx1250) — compile-verified
//
#include <hip/hip_runtime.h>
#include <hip/hip_bf16.h>

typedef __attribute__((ext_vector_type(2))) float v2f;
typedef __attribute__((ext_vector_type(8))) float v8f;

#define T_FR 16
#define SH 480
#define SW 640
#define IH 384
#define IW 512
#define NPIX (IH * IW)            // 196608
#define NSTRIP 30                 // 480 / 16
#define NKSEG 4                   // K split: 4 segments of 160 columns
#define KSEGW 160
#define NSTEP (KSEGW / 4)         // 40 wmma steps per segment
#define PB 8                      // pipeline batch (8 wmma steps)
#define NBATCH (NSTEP / PB)       // 5 batches
#define NTASK (48 * NSTRIP * NKSEG) // 5760 wave tasks
#define NPART (NSTRIP * NKSEG)    // 120 partials per (t,c)

// workspace layout (floats)
#define WS_WCOL 0                 // 640
#define WS_WROW 640               // 480
#define WS_S    1120              // 48 * 120 strip/kseg partials
#define WS_DX   6880              // 16
#define WS_DY   6896              // 16
#define WS_FM   6912              // 16 (total 6928 floats)

// output offsets (floats)
#define O_SPARSE 0
#define O_XY     393216
#define O_Z      6684672
#define O_VIS    9830400
#define O_TRAJ   12976128

// ---------------------------------------------------------------------------
// Resize weights: wcol[x] = sum of bilinear (align_corners) weights that source
// column x receives over all 512 output columns; wrow likewise over 384 rows.
// ---------------------------------------------------------------------------
__global__ void k_weights(float* __restrict__ ws) {
  int i = blockIdx.x * blockDim.x + threadIdx.x;
  if (i < SW) {
    float w = 0.f;
    int c0 = (int)floorf((float)i * 511.0f / 639.0f) - 2;
    for (int k = 0; k < 5; ++k) {
      int o = c0 + k;
      if (o < 0 || o >= IW) continue;
      float xs = ((float)o * 639.0f) / 511.0f;
      int x0 = (int)floorf(xs);
      float f = xs - (float)x0;
      int x1 = min(x0 + 1, SW - 1);
      if (x0 == i) w += 1.0f - f;
      if (x1 == i) w += f;
    }
    ws[WS_WCOL + i] = w;
  } else if (i < SW + SH) {
    int j = i - SW;
    float w = 0.f;
    int c0 = (int)floorf((float)j * 383.0f / 479.0f) - 2;
    for (int k = 0; k < 5; ++k) {
      int o = c0 + k;
      if (o < 0 || o >= IH) continue;
      float ys = ((float)o * 479.0f) / 383.0f;
      int y0 = (int)floorf(ys);
      float f = ys - (float)y0;
      int y1 = min(y0 + 1, SH - 1);
      if (y0 == j) w += 1.0f - f;
      if (y1 == j) w += f;
    }
    ws[WS_WROW + j] = w;
  }
}

// ---------------------------------------------------------------------------
// feat reduction via WMMA: each wave handles one (t,c,strip16,kseg160).
// Batched software pipeline: load 16x b64 for batch k+1, then 8 chained
// V_WMMA_F32_16X16X4_F32 on batch k. Fully unrolled -> no register rotation,
// loads a full batch ahead of consumption.
// A 16x4 layout: lane l<16 -> M=l, K={0,1}; lane>=16 -> M=l-16, K={2,3}.
// ---------------------------------------------------------------------------
__global__ void k_feat(const float* __restrict__ video, float* __restrict__ ws) {
  int wid  = blockIdx.x * (blockDim.x >> 5) + (threadIdx.x >> 5);
  int lane = threadIdx.x & 31;
  if (wid >= NTASK) return;                 // uniform per wave (exact grid anyway)
  int tc    = wid / NPART;                  // (t*3 + c), 0..47
  int r     = wid - tc * NPART;
  int strip = r >> 2;                       // 0..29
  int kseg  = r & 3;                        // 0..3
  int yb    = strip * 16;
  int kb0   = kseg * KSEGW;

  const float* base = video + (size_t)tc * SH * SW;
  const float* wcol = ws + WS_WCOL;
  const float* wrow = ws + WS_WROW;

  int half = lane >> 4;                     // 0: K 0..1, 1: K 2..3
  int rowl = lane & 15;
  const float* rp = base + (size_t)(yb + rowl) * SW + half * 2;
  const float* bp = wcol + half * 2;

  v2f A[PB], B[PB];
#pragma unroll
  for (int p = 0; p < PB; ++p) {            // prologue: batch 0
    int kb = kb0 + p * 4;
    A[p].x = rp[kb]; A[p].y = rp[kb + 1];
    B[p].x = bp[kb]; B[p].y = bp[kb + 1];
  }

  v8f acc = {0.f, 0.f, 0.f, 0.f, 0.f, 0.f, 0.f, 0.f};
#pragma unroll
  for (int blk = 0; blk < NBATCH - 1; ++blk) {
    int kn = kb0 + (blk + 1) * PB * 4;
    v2f NA[PB], NB[PB];
    __builtin_prefetch(rp + kn + PB * 8, 0, 0);   // one speculative prefetch/batch
#pragma unroll
    for (int p = 0; p < PB; ++p) {          // issue batch blk+1 loads
      int kb = kn + p * 4;
      NA[p].x = rp[kb]; NA[p].y = rp[kb + 1];
      NB[p].x = bp[kb]; NB[p].y = bp[kb + 1];
    }
#pragma unroll
    for (int p = 0; p < PB; ++p)            // consume batch blk
      acc = __builtin_amdgcn_wmma_f32_16x16x4_f32(
          false, A[p], false, B[p], (short)0, acc, false, false);
#pragma unroll
    for (int p = 0; p < PB; ++p) { A[p] = NA[p]; B[p] = NB[p]; }
  }
#pragma unroll
  for (int p = 0; p < PB; ++p)              // epilogue: last batch
    acc = __builtin_amdgcn_wmma_f32_16x16x4_f32(
        false, A[p], false, B[p], (short)0, acc, false, false);

  // lanes 0-15 hold U[0..7] in acc, lanes 16-31 hold U[8..15]
  float s = 0.f;
#pragma unroll
  for (int q = 0; q < 8; ++q) s += acc[q] * wrow[yb + half * 8 + q];
  s += __shfl_xor(s, 16, 32);               // combine the two halves
  if (lane == 0) ws[WS_S + tc * NPART + strip * 4 + kseg] = s;
}

// deterministic finish: sum 120 partials per (t,c), derive disp and feat-mean
__global__ void k_prep(float* __restrict__ ws) {
  int t = threadIdx.x;
  if (t >= T_FR) return;
  float f[3];
  for (int c = 0; c < 3; ++c) {
    const float* p = ws + WS_S + (t * 3 + c) * NPART;
    float s = 0.f;
    for (int k = 0; k < NPART; ++k) s += p[k];
    f[c] = s * (1.0f / (float)NPIX);
  }
  ws[WS_DX + t] = tanhf(f[0]) * 2.0f;
  ws[WS_DY + t] = tanhf(f[1]) * 2.0f;
  ws[WS_FM + t] = (f[0] + f[1] + f[2]) * (1.0f / 3.0f);
}

// ---------------------------------------------------------------------------
// Fused per-pixel kernel: one thread per (t, y, x), produces all 5 outputs.
// ---------------------------------------------------------------------------
__global__ void k_main(const float* __restrict__ video,
                       const float* __restrict__ depth,
                       const float* __restrict__ ws,
                       float* __restrict__ out) {
  int idx = blockIdx.x * blockDim.x + threadIdx.x;
  int t   = idx / NPIX;                     // uniform per block (NPIX % 256 == 0)
  int rem = idx - t * NPIX;
  int y   = rem >> 9;                       // IW == 512
  int x   = rem & 511;

  float dx = ws[WS_DX + t], dy = ws[WS_DY + t], fm = ws[WS_FM + t];

  float trackx = 0.f, tracky = 0.f, zval = 0.f, visb = 0.f;
  if (x < 510) {                            // query grid covers x in [0,510)
    float tf = (float)t;
    float tx = fminf(fmaxf((float)x + tf * dx, 0.f), 511.f);
    float ty = fminf(fmaxf((float)y + tf * dy, 0.f), 383.f);
    int x0 = (int)floorf(tx); int x1 = min(x0 + 1, 511);
    int y0 = (int)floorf(ty); int y1 = min(y0 + 1, 383);
    float wx = tx - (float)x0, wy = ty - (float)y0;
    // nearest-resized depth at integer (xi,yi): depth[floor(yi*1.25)][floor(xi*1.25)]
    const float* dep = depth + (size_t)t * SH * SW;
    int ix0 = (x0 * 5) >> 2, ix1 = (x1 * 5) >> 2;
    int iy0 = (y0 * 5) >> 2, iy1 = (y1 * 5) >> 2;
    float d00 = dep[iy0 * SW + ix0], d01 = dep[iy0 * SW + ix1];
    float d10 = dep[iy1 * SW + ix0], d11 = dep[iy1 * SW + ix1];
    zval = d00 * (1.f - wx) * (1.f - wy) + d01 * wx * (1.f - wy)
         + d10 * (1.f - wx) * wy         + d11 * wx * wy;
    trackx = tx * (639.0f / 511.0f);
    tracky = ty * (479.0f / 383.0f);
    float vv  = 3.0f * sinf(0.05f * (float)(x + y)) + fm;
    float sig = 1.0f / (1.0f + expf(-vv));
    visb = (sig > 0.8f) ? 1.f : 0.f;
  }

  // trajs_3d: X=(x-320)/640, Y=(y-240)/640, Z=depth, color via grid_sample
  float X = (trackx - 320.0f) * (1.0f / 640.0f);
  float Y = (tracky - 240.0f) * (1.0f / 640.0f);
  float px = fminf(fmaxf(trackx * (639.0f / 640.0f), 0.f), 639.f);
  float py = fminf(fmaxf(tracky * (479.0f / 480.0f), 0.f), 479.f);
  int cx0 = (int)floorf(px); int cx1 = min(cx0 + 1, 639);
  int cy0 = (int)floorf(py); int cy1 = min(cy0 + 1, 479);
  float cwx = px - (float)cx0, cwy = py - (float)cy0;
  float w00 = (1.f - cwx) * (1.f - cwy), w01 = cwx * (1.f - cwy);
  float w10 = (1.f - cwx) * cwy,         w11 = cwx * cwy;
  const float* vt = video + (size_t)t * 3 * SH * SW;
  float col[3];
#pragma unroll
  for (int c = 0; c < 3; ++c) {
    const float* p = vt + c * SH * SW;
    col[c] = p[cy0 * SW + cx0] * w00 + p[cy0 * SW + cx1] * w01
           + p[cy1 * SW + cx0] * w10 + p[cy1 * SW + cx1] * w11;
  }

  // ---- writes ----
  out[O_XY + idx * 2 + 0] = trackx;
  out[O_XY + idx * 2 + 1] = tracky;
  out[O_Z  + idx] = zval;
  out[O_VIS + idx] = visb;
  int tb = O_TRAJ + idx * 7;
  out[tb + 0] = X;      out[tb + 1] = Y;      out[tb + 2] = zval;
  out[tb + 3] = col[0]; out[tb + 4] = col[1]; out[tb + 5] = col[2];
  out[tb + 6] = visb;
  if (((x & 3) == 0) && ((y & 3) == 0)) {
    int m = t * 12288 + (y >> 2) * 128 + (x >> 2);
    out[O_SPARSE + m * 2 + 0] = trackx;
    out[O_SPARSE + m * 2 + 1] = tracky;
  }
}

extern "C" void kernel_launch(void* const* d_in, const int* in_sizes, int n_in,
                              void* d_out, int out_size, void* d_ws, size_t ws_size,
                              hipStream_t stream) {
  const float* video = (const float*)d_in[0];  // (1,16,3,480,640) f32
  const float* depth = (const float*)d_in[1];  // (1,16,1,480,640) f32
  float* out = (float*)d_out;
  float* ws  = (float*)d_ws;

  k_weights<<<5, 256, 0, stream>>>(ws);                    // 640+480 weights
  k_feat<<<NTASK / 8, 256, 0, stream>>>(video, ws);        // 5760 waves, WMMA
  k_prep<<<1, 32, 0, stream>>>(ws);                        // disp / feat-mean
  k_main<<<(T_FR * NPIX) / 256, 256, 0, stream>>>(video, depth, ws, out);
}